// DetectionOutput_27384711479419
// MI455X (gfx1250) — compile-verified
//
#include <hip/hip_runtime.h>
#include <cstdint>

#define NUM_CLASSES 81
#define TOPK 200
#define CONF_THRV 0.01f
#define NMS_THRV 0.45f
#define BATCH 32
#define PRIORS 24564
#define NTILE 13           // ceil(208/16)
#define PAD 208            // padded candidate count for tiling
#define N3 (NUM_CLASSES * TOPK)   // 16200

typedef __attribute__((ext_vector_type(2))) float v2f;
typedef __attribute__((ext_vector_type(8))) float v8f;

// Monotone 16-bit key: descending float order == descending key order.
__device__ __forceinline__ unsigned short key16_of(float f) {
  unsigned u = __float_as_uint(f);
  u = (u & 0x80000000u) ? ~u : (u | 0x80000000u);
  return (unsigned short)(u >> 16);
}

// ---------------- kernel 1: box decode (bandwidth bound) ----------------
__global__ __launch_bounds__(256) void decode_kernel(
    const float* __restrict__ loc, const float* __restrict__ prior,
    float* __restrict__ dec) {
  int idx = blockIdx.x * 256 + threadIdx.x;
  const int total = BATCH * PRIORS;
  if (idx >= total) return;
  int p = idx % PRIORS;
  const float* l  = loc + (size_t)idx * 4;
  const float* pb = prior + (size_t)p * 4;
  const float* vr = prior + (size_t)PRIORS * 4 + (size_t)p * 4;
  float pw = pb[2] - pb[0], ph = pb[3] - pb[1];
  float pcx = (pb[0] + pb[2]) * 0.5f, pcy = (pb[1] + pb[3]) * 0.5f;
  float cx = vr[0] * l[0] * pw + pcx;
  float cy = vr[1] * l[1] * ph + pcy;
  float w  = expf(vr[2] * l[2]) * pw;
  float h  = expf(vr[3] * l[3]) * ph;
  float* o = dec + (size_t)idx * 4;
  o[0] = cx - w * 0.5f; o[1] = cy - h * 0.5f;
  o[2] = cx + w * 0.5f; o[3] = cy + h * 0.5f;
}

// ------------- shared routine: radix-select top-200 + sort 256 -------------
// key16[] must be pre-populated. Selects top-TOPK by 16-bit key (2x 8-bit
// radix passes), gathers exact scores from global, bitonic-sorts 256 slots
// descending. Results in cscore/cidx (first TOPK valid).
__device__ void topk_select_sort(const float* gbase, int gstride, int N,
                                 int applyThr, unsigned short* key16,
                                 unsigned* hist, unsigned* ctl,
                                 float* cscore, unsigned* cidx) {
  int tid = threadIdx.x;
  // ---- pass 1: high byte ----
  hist[tid] = 0u; __syncthreads();
  for (int i = tid; i < N; i += 256) atomicAdd(&hist[key16[i] >> 8], 1u);
  __syncthreads();
  if (tid == 0) {
    unsigned r = TOPK, cum = 0, sel = 0;
    for (int bb = 255; bb >= 0; --bb) {
      unsigned h = hist[bb];
      if (cum + h >= r) { sel = (unsigned)bb; r = r - cum; break; }
      cum += h;
    }
    ctl[0] = sel; ctl[1] = r;
  }
  __syncthreads();
  unsigned selHi = ctl[0];
  hist[tid] = 0u; __syncthreads();
  // ---- pass 2: low byte within selected bin ----
  for (int i = tid; i < N; i += 256) {
    unsigned k = key16[i];
    if ((k >> 8) == selHi) atomicAdd(&hist[k & 255u], 1u);
  }
  __syncthreads();
  if (tid == 0) {
    unsigned r = ctl[1], cum = 0, sel = 0;
    for (int bb = 255; bb >= 0; --bb) {
      unsigned h = hist[bb];
      if (cum + h >= r) { sel = (unsigned)bb; break; }
      cum += h;
    }
    ctl[2] = sel; ctl[3] = 0u;
  }
  __syncthreads();
  unsigned kth = (selHi << 8) | ctl[2];
  // ---- gather strictly greater (count < TOPK) ----
  for (int i = tid; i < N; i += 256) {
    if (key16[i] > kth) {
      unsigned pos = atomicAdd(&ctl[3], 1u);
      if (pos < 256u) {
        float s = gbase[(size_t)i * gstride];
        if (applyThr) s = (s > CONF_THRV) ? s : -1.0f;
        cscore[pos] = s; cidx[pos] = (unsigned)i;
      }
    }
  }
  __syncthreads();
  // ---- gather ties at the boundary (capped) ----
  for (int i = tid; i < N; i += 256) {
    if (key16[i] == kth) {
      unsigned pos = atomicAdd(&ctl[3], 1u);
      if (pos < 256u) {
        float s = gbase[(size_t)i * gstride];
        if (applyThr) s = (s > CONF_THRV) ? s : -1.0f;
        cscore[pos] = s; cidx[pos] = (unsigned)i;
      }
    }
  }
  __syncthreads();
  unsigned cnt = ctl[3]; if (cnt > 256u) cnt = 256u;
  if ((unsigned)tid >= cnt) { cscore[tid] = -3.0e38f; cidx[tid] = 0u; }
  __syncthreads();
  // ---- bitonic sort 256, descending by score ----
  for (int k = 2; k <= 256; k <<= 1) {
    for (int j = k >> 1; j > 0; j >>= 1) {
      int ixj = tid ^ j;
      if (ixj > tid) {
        float a = cscore[tid], bv = cscore[ixj];
        bool desc = ((tid & k) == 0);
        if (desc ? (a < bv) : (a > bv)) {
          cscore[tid] = bv; cscore[ixj] = a;
          unsigned t = cidx[tid]; cidx[tid] = cidx[ixj]; cidx[ixj] = t;
        }
      }
      __syncthreads();
    }
  }
}

// --------- kernel 2: per-(batch,class) top-K + NMS (one block each) ---------
__global__ __launch_bounds__(256) void nms_kernel(
    const float* __restrict__ conf, const float* __restrict__ dec,
    float* __restrict__ nscores, float* __restrict__ nboxes) {
  __shared__ union __align__(16) {
    struct { unsigned short key16[24576]; float stage[2][1024]; } p1; // 57344 B
    struct { unsigned adj[PAD * 8]; float lbox[PAD][4]; float area[PAD]; } p2;
  } su;
  __shared__ float    cscore[256];
  __shared__ unsigned cidx[256];
  __shared__ unsigned hist[256];
  __shared__ unsigned ctl[4];
  __shared__ unsigned sup[8];
  __shared__ unsigned kb[8];

  int tid = threadIdx.x;
  int b = blockIdx.x / NUM_CLASSES;
  int c = blockIdx.x % NUM_CLASSES;
  size_t obase = ((size_t)b * NUM_CLASSES + c) * TOPK;

  if (c == 0) {  // background class: reference forces score -1 -> nothing kept
    for (int k = tid; k < TOPK; k += 256) {
      nscores[obase + k] = 0.f;
      #pragma unroll
      for (int j = 0; j < 4; ++j) nboxes[(obase + k) * 4 + j] = 0.f;
    }
    return;
  }

  const float* gbase = conf + (size_t)b * PRIORS * NUM_CLASSES + c; // stride 81

  // ---- phase 1: async-pipelined strided gather -> thresholded 16-bit keys
  auto issue = [&](int ch, int buf) {
    #pragma unroll
    for (int k = 0; k < 4; ++k) {
      int e = ch * 1024 + k * 256 + tid;
      if (e < PRIORS) {
        unsigned lds = (unsigned)(size_t)&su.p1.stage[buf][k * 256 + tid];
        const float* g = gbase + (size_t)e * NUM_CLASSES;
        asm volatile("global_load_async_to_lds_b32 %0, %1, off"
                     :: "v"(lds), "v"(g) : "memory");
      }
    }
  };
  auto convert = [&](int ch, int buf) {
    #pragma unroll
    for (int k = 0; k < 4; ++k) {
      int e = ch * 1024 + k * 256 + tid;
      if (e < PRIORS) {
        float f = su.p1.stage[buf][k * 256 + tid];
        float t = (f > CONF_THRV) ? f : -1.0f;
        su.p1.key16[e] = key16_of(t);
      }
    }
  };
  const int CH = (PRIORS + 1023) / 1024;  // 24
  issue(0, 0);
  for (int ch = 0; ch < CH; ++ch) {
    if (ch + 1 < CH) {
      issue(ch + 1, (ch + 1) & 1);
      asm volatile("s_wait_asynccnt 4" ::: "memory"); // older chunk complete
    } else {
      asm volatile("s_wait_asynccnt 0" ::: "memory");
    }
    convert(ch, ch & 1);  // each wave reads only what it loaded
  }
  __syncthreads();

  // ---- phase 2: select + sort top-200 candidates
  topk_select_sort(gbase, NUM_CLASSES, PRIORS, 1, su.p1.key16, hist, ctl,
                   cscore, cidx);
  __syncthreads();

  // ---- phase 3: init adjacency, async-gather candidate boxes, areas
  for (int i = tid; i < PAD * 8; i += 256) su.p2.adj[i] = 0u;
  for (int i = tid; i < PAD; i += 256) {
    su.p2.area[i] = 0.f;
    su.p2.lbox[i][0] = su.p2.lbox[i][1] = su.p2.lbox[i][2] = su.p2.lbox[i][3] = 0.f;
  }
  if (tid < 8) { sup[tid] = 0u; kb[tid] = 0u; }
  __syncthreads();
  if (tid < TOPK) {
    unsigned pi = cidx[tid];
    const float* g = dec + ((size_t)b * PRIORS + pi) * 4;
    unsigned lds = (unsigned)(size_t)&su.p2.lbox[tid][0];
    asm volatile("global_load_async_to_lds_b128 %0, %1, off"
                 :: "v"(lds), "v"(g) : "memory");
  }
  asm volatile("s_wait_asynccnt 0" ::: "memory");
  __syncthreads();
  if (tid < TOPK) {
    float x1 = su.p2.lbox[tid][0], y1 = su.p2.lbox[tid][1];
    float x2 = su.p2.lbox[tid][2], y2 = su.p2.lbox[tid][3];
    su.p2.area[tid] = fmaxf(x2 - x1, 0.f) * fmaxf(y2 - y1, 0.f);
  }
  __syncthreads();

  // ---- phase 4: pairwise IoU in 16x16 tiles; union via V_WMMA_F32_16X16X4_F32
  // D = A(16x4) x B(4x16) + C, A row = [area_i, 1, 0, 0], B col = [1, area_j, 0, 0]
  // => D[M][N] = area_i[M] + area_j[N] - inter[M][N]  (C = -inter)
  // Threshold test uses inter > thr*union (divisor > 0) -> no division.
  int wave = tid >> 5, lane = tid & 31;
  int half = lane >> 4, l15 = lane & 15;
  for (int t = wave; t < NTILE * NTILE; t += 8) {   // wave-uniform loop
    int ti = t / NTILE, tj = t % NTILE;
    int gj = tj * 16 + l15;                         // column index (both halves)
    float jx1 = su.p2.lbox[gj][0], jy1 = su.p2.lbox[gj][1];
    float jx2 = su.p2.lbox[gj][2], jy2 = su.p2.lbox[gj][3];
    // A layout: VGPR0 = K0 (lanes 0-15) / K2 (16-31); VGPR1 = K1 / K3
    v2f a, bm;
    a.x  = half ? 0.f : su.p2.area[ti * 16 + l15];  // K0 = area_i
    a.y  = half ? 0.f : 1.0f;                       // K1 = 1
    bm.x = half ? 0.f : 1.0f;                       // row K0 = ones
    bm.y = half ? 0.f : su.p2.area[gj];             // row K1 = area_j
    float inter[8];
    v8f cacc;
    #pragma unroll
    for (int r = 0; r < 8; ++r) {
      int gi = ti * 16 + r + half * 8;              // D layout: M = r / r+8
      float ix1 = su.p2.lbox[gi][0], iy1 = su.p2.lbox[gi][1];
      float ix2 = su.p2.lbox[gi][2], iy2 = su.p2.lbox[gi][3];
      float xx1 = fmaxf(ix1, jx1), yy1 = fmaxf(iy1, jy1);
      float xx2 = fminf(ix2, jx2), yy2 = fminf(iy2, jy2);
      float it = fmaxf(xx2 - xx1, 0.f) * fmaxf(yy2 - yy1, 0.f);
      inter[r] = it;
      cacc[r] = -it;
    }
    v8f uni = __builtin_amdgcn_wmma_f32_16x16x4_f32(
        false, a, false, bm, (short)0, cacc, false, false);
    // Pack each row's 16 predicate bits with a wave32 ballot (full EXEC):
    //   bits [15:0]  -> row ti*16+r   (lanes 0..15)
    //   bits [31:16] -> row ti*16+r+8 (lanes 16..31)
    // A 16-aligned column window lies inside one 32-bit adjacency word:
    //   word = tj>>1, shift = (tj&1)*16.
    unsigned mrow[8];
    #pragma unroll
    for (int r = 0; r < 8; ++r) {
      bool pred = inter[r] > NMS_THRV * fmaxf(uni[r], 1e-10f);
      mrow[r] = __builtin_amdgcn_ballot_w32(pred);
    }
    unsigned wsel = (unsigned)(tj >> 1);
    unsigned wsh  = (unsigned)((tj & 1) * 16);
    if (l15 == 0) {  // single divergent region: lanes 0 and 16 write 8 rows
      int rowbase = ti * 16 + half * 8;
      #pragma unroll
      for (int r = 0; r < 8; ++r) {
        unsigned bits = (half ? (mrow[r] >> 16) : (mrow[r] & 0xFFFFu)) << wsh;
        atomicOr(&su.p2.adj[(rowbase + r) * 8 + wsel], bits);
      }
    }
  }
  __syncthreads();

  // ---- phase 5: greedy suppression (wave 0, bitmask words on lanes 0-7)
  if (wave == 0) {
    for (int i = 0; i < TOPK; ++i) {
      unsigned w = (unsigned)i >> 5, m = 1u << (i & 31);
      bool suppressed = (sup[w] & m) != 0u;
      bool kept = (cscore[i] > CONF_THRV) && !suppressed;
      if (kept) {
        if (lane < 8) sup[lane] |= su.p2.adj[i * 8 + lane];
        if (lane == 0) kb[w] |= m;
      }
    }
  }
  __syncthreads();
  for (int k = tid; k < TOPK; k += 256) {
    bool kept = (kb[k >> 5] >> (k & 31)) & 1u;
    nscores[obase + k] = kept ? cscore[k] : 0.f;
    #pragma unroll
    for (int j = 0; j < 4; ++j)
      nboxes[(obase + k) * 4 + j] = su.p2.lbox[k][j];
  }
}

// --------- kernel 3: per-batch final top-200 across 81*200 detections ---------
__global__ __launch_bounds__(256) void finalize_kernel(
    const float* __restrict__ nscores, const float* __restrict__ nboxes,
    float* __restrict__ out) {
  __shared__ unsigned short key16[16384];
  __shared__ float    cscore[256];
  __shared__ unsigned cidx[256];
  __shared__ unsigned hist[256];
  __shared__ unsigned ctl[4];
  int tid = threadIdx.x;
  int b = blockIdx.x;
  const float* gbase = nscores + (size_t)b * N3;
  for (int i = tid; i < N3; i += 256) key16[i] = key16_of(gbase[i]);
  __syncthreads();
  topk_select_sort(gbase, 1, N3, 0, key16, hist, ctl, cscore, cidx);
  __syncthreads();
  if (tid < TOPK) {
    float s = cscore[tid];
    unsigned fi = cidx[tid];
    float* row = out + ((size_t)b * TOPK + tid) * 7;
    if (s > 0.f) {
      row[0] = (float)b;
      row[1] = (float)(fi / TOPK);
      row[2] = s;
      const float* bx = nboxes + ((size_t)b * N3 + fi) * 4;
      row[3] = bx[0]; row[4] = bx[1]; row[5] = bx[2]; row[6] = bx[3];
    } else {
      #pragma unroll
      for (int j = 0; j < 7; ++j) row[j] = 0.f;
    }
  }
}

extern "C" void kernel_launch(void* const* d_in, const int* in_sizes, int n_in,
                              void* d_out, int out_size, void* d_ws, size_t ws_size,
                              hipStream_t stream) {
  const float* loc   = (const float*)d_in[0];
  const float* conf  = (const float*)d_in[1];
  const float* prior = (const float*)d_in[2];
  float* out = (float*)d_out;

  float* dec     = (float*)d_ws;                                   // B*P*4
  float* nscores = dec + (size_t)BATCH * PRIORS * 4;               // B*C*200
  float* nboxes  = nscores + (size_t)BATCH * NUM_CLASSES * TOPK;   // B*C*200*4

  int total = BATCH * PRIORS;
  decode_kernel<<<(total + 255) / 256, 256, 0, stream>>>(loc, prior, dec);
  nms_kernel<<<BATCH * NUM_CLASSES, 256, 0, stream>>>(conf, dec, nscores, nboxes);
  finalize_kernel<<<BATCH, 256, 0, stream>>>(nscores, nboxes, out);
}